// PPLoss_18580028523148
// MI455X (gfx1250) — compile-verified
//
#include <hip/hip_runtime.h>
#include <math.h>

typedef __attribute__((ext_vector_type(2))) float v2f;
typedef __attribute__((ext_vector_type(8))) float v8f;

#define NC 256   // classes
#define DD 512   // feature dim
#define NB 512   // queries
#define NS 8192  // support samples

// workspace layout in floats
#define OFF_MUS    0         // 256*512  class sums over xs
#define OFF_SUMSQ  131072    // 256*512  class sums over xq
#define OFF_P      262144    // 256*512  prototypes mus/max(cc,0.1)
#define OFF_LOGIT  393216    // 512*256
#define OFF_CC     524288    // 256  class_count(ys)
#define OFF_CQ     524544    // 256  counts(yq)
#define OFF_PNORM  524800    // 256
#define OFF_QNORM  525056    // 512
#define OFF_LOSSR  525568    // 512
#define OFF_D2R    526080    // 512

__device__ __forceinline__ float blk_sum(float v, float* sm) {
    int t = threadIdx.x;
    sm[t] = v; __syncthreads();
    for (int s = 128; s > 0; s >>= 1) {
        if (t < s) sm[t] += sm[t + s];
        __syncthreads();
    }
    float r = sm[0];
    __syncthreads();
    return r;
}

__global__ void k_zero(float* p, int n) {
    int i = blockIdx.x * blockDim.x + threadIdx.x;
    if (i < n) p[i] = 0.0f;
}

// histogram of labels -> float counts (one block, 256 threads)
__global__ void k_count(const int* __restrict__ y, int n, float* __restrict__ out) {
    __shared__ int cnt[NC];
    int t = threadIdx.x;
    cnt[t] = 0;
    __syncthreads();
    for (int i = t; i < n; i += 256) atomicAdd(&cnt[y[i]], 1);
    __syncthreads();
    out[t] = (float)cnt[t];
}

// per-class feature sums; each block owns a 32-dim slice (grid = 16), LDS-accumulated
__global__ void k_segsum(const float* __restrict__ x, const int* __restrict__ y,
                         int n, float* __restrict__ out) {
    __shared__ float acc[NC * 32];
    int t = threadIdx.x;
    int dg = blockIdx.x;               // 16 groups of 32 dims
    for (int i = t; i < NC * 32; i += 256) acc[i] = 0.0f;
    __syncthreads();
    int d  = t & 31;
    int sl = t >> 5;                   // 8 samples in flight
    for (int s0 = 0; s0 < n; s0 += 8) {
        int s = s0 + sl;
        int cls = y[s];
        float v = x[s * DD + dg * 32 + d];
        atomicAdd(&acc[cls * 32 + d], v);   // ds_add_f32
    }
    __syncthreads();
    for (int i = t; i < NC * 32; i += 256) {
        int c  = i >> 5;
        int dd = i & 31;
        out[c * DD + dg * 32 + dd] = acc[i];
    }
}

// P[c,:] = mus[c,:]/max(cc,0.1); pnorm[c] = ||P[c]||^2  (one block per class)
__global__ void k_proto(const float* __restrict__ mus, const float* __restrict__ cc,
                        float* __restrict__ P, float* __restrict__ pnorm) {
    __shared__ float sm[256];
    int c = blockIdx.x, t = threadIdx.x;
    float denom = fmaxf(cc[c], 0.1f);
    float acc = 0.0f;
    for (int d = t; d < DD; d += 256) {
        float p = mus[c * DD + d] / denom;
        P[c * DD + d] = p;
        acc += p * p;
    }
    float s = blk_sum(acc, sm);
    if (t == 0) pnorm[c] = s;
}

// ||row||^2 for a [R,512] matrix (one block per row)
__global__ void k_rownorm(const float* __restrict__ x, float* __restrict__ out) {
    __shared__ float sm[256];
    int r = blockIdx.x, t = threadIdx.x;
    float acc = 0.0f;
    for (int d = t; d < DD; d += 256) { float v = x[r * DD + d]; acc += v * v; }
    float s = blk_sum(acc, sm);
    if (t == 0) out[r] = s;
}

// intra-class variance pieces: d2r[i] = (count>=3) ? ||xq[i]-mean||^2 : 0
__global__ void k_var(const float* __restrict__ xq, const int* __restrict__ yq,
                      const float* __restrict__ sums, const float* __restrict__ cq,
                      float* __restrict__ d2r) {
    __shared__ float sm[256];
    int i = blockIdx.x, t = threadIdx.x;
    int cls = yq[i];
    float cnt = cq[cls];
    float denom = fmaxf(cnt, 1.0f);
    float acc = 0.0f;
    for (int d = t; d < DD; d += 256) {
        float diff = xq[i * DD + d] - sums[cls * DD + d] / denom;
        acc += diff * diff;
    }
    float s = blk_sum(acc, sm);
    if (t == 0) d2r[i] = (cnt >= 3.0f) ? s : 0.0f;
}

// G = xq @ P^T via f32 WMMA; logit = valid * -0.5*(qn - 2G + pn)
// one wave per 16x16 tile: 32 B-tiles x 16 C-tiles = 512 tiles = 64 blocks x 8 waves
__global__ void k_wmma_logit(const float* __restrict__ xq, const float* __restrict__ P,
                             const float* __restrict__ pnorm, const float* __restrict__ qnorm,
                             const float* __restrict__ cc, float* __restrict__ logit) {
    int wave = threadIdx.x >> 5;
    int lane = threadIdx.x & 31;
    int tile = blockIdx.x * 8 + wave;
    int bm = tile >> 4;       // 0..31 query tile
    int cn = tile & 15;       // 0..15 class tile
    int mn = lane & 15;
    int kb = (lane >> 4) << 1;   // K base: 0 or 2
    const float* arow = xq + (bm * 16 + mn) * DD + kb;   // A: M=lane%16, K=kb+{0,1}
    const float* brow = P  + (cn * 16 + mn) * DD + kb;   // B: N=lane%16 (P row-major over dims)
    v8f acc = {};
    for (int k = 0; k < DD; k += 4) {
        v2f a = *(const v2f*)(arow + k);
        v2f b = *(const v2f*)(brow + k);
        acc = __builtin_amdgcn_wmma_f32_16x16x4_f32(
            /*neg_a=*/false, a, /*neg_b=*/false, b,
            /*c_mod=*/(short)0, acc, /*reuse_a=*/false, /*reuse_b=*/false);
    }
    int n  = cn * 16 + (lane & 15);
    float pn  = pnorm[n];
    float vld = (cc[n] > 0.1f) ? 1.0f : 0.0f;
    int mbase = bm * 16 + ((lane >> 4) << 3);
    for (int j = 0; j < 8; j++) {
        int m = mbase + j;                       // C/D layout: M = j + 8*(lane/16)
        float l = -0.5f * (qnorm[m] - 2.0f * acc[j] + pn);
        logit[m * NC + n] = l * vld;
    }
}

// leave-one-out fixup at c == yq_new[b]
__global__ void k_fixup(const float* __restrict__ xq, const float* __restrict__ mus,
                        const float* __restrict__ cc, const int* __restrict__ ys,
                        const int* __restrict__ pos, float* __restrict__ logit) {
    __shared__ float sm[256];
    int b = blockIdx.x, t = threadIdx.x;
    int yb = ys[pos[b]];
    float cn = cc[yb] - 1.0f;
    float denom = fmaxf(cn, 0.1f);
    float acc = 0.0f;
    for (int d = t; d < DD; d += 256) {
        float x = xq[b * DD + d];
        float proto = (mus[yb * DD + d] - x) / denom;
        float diff = x - proto;
        acc += diff * diff;
    }
    float s = blk_sum(acc, sm);
    if (t == 0) logit[b * NC + yb] = (cn > 0.1f) ? (-0.5f * s) : 0.0f;
}

// per-row log-softmax NLL (one block per query, one class per thread)
__global__ void k_loss(const float* __restrict__ logit, const int* __restrict__ ys,
                       const int* __restrict__ pos, float* __restrict__ lossr) {
    __shared__ float sm[256];
    int b = blockIdx.x, t = threadIdx.x;
    float l = logit[b * NC + t];
    sm[t] = l; __syncthreads();
    for (int s = 128; s > 0; s >>= 1) {
        if (t < s) sm[t] = fmaxf(sm[t], sm[t + s]);
        __syncthreads();
    }
    float mx = sm[0]; __syncthreads();
    float se = blk_sum(__expf(l - mx), sm);
    if (t == 0) {
        int yb = ys[pos[b]];
        float lt = logit[b * NC + yb];
        lossr[b] = -(lt - mx - logf(se));
    }
}

__global__ void k_final(const float* __restrict__ lossr, const float* __restrict__ d2r,
                        const float* __restrict__ cq, float* __restrict__ out) {
    __shared__ float sm[256];
    int t = threadIdx.x;
    float loss = blk_sum(lossr[t] + lossr[t + 256], sm) * (1.0f / 512.0f);
    float tot  = blk_sum(d2r[t] + d2r[t + 256], sm);
    float cv   = cq[t];
    float cnt  = blk_sum((cv >= 3.0f) ? cv : 0.0f, sm);
    if (t == 0) {
        out[0] = loss;
        out[1] = (cnt > 0.0f) ? (tot / fmaxf(cnt, 1.0f)) : 0.0f;
    }
}

extern "C" void kernel_launch(void* const* d_in, const int* in_sizes, int n_in,
                              void* d_out, int out_size, void* d_ws, size_t ws_size,
                              hipStream_t stream) {
    const float* xq  = (const float*)d_in[0];
    const int*   yq  = (const int*)d_in[1];
    const float* xs  = (const float*)d_in[2];
    const int*   ys  = (const int*)d_in[3];
    const int*   pos = (const int*)d_in[4];
    float* out = (float*)d_out;
    float* ws  = (float*)d_ws;

    // zero the two accumulation regions (mus + sums_q)
    k_zero<<<(262144 + 255) / 256, 256, 0, stream>>>(ws + OFF_MUS, 262144);

    k_count<<<1, 256, 0, stream>>>(ys, NS, ws + OFF_CC);
    k_count<<<1, 256, 0, stream>>>(yq, NB, ws + OFF_CQ);

    k_segsum<<<16, 256, 0, stream>>>(xs, ys, NS, ws + OFF_MUS);
    k_segsum<<<16, 256, 0, stream>>>(xq, yq, NB, ws + OFF_SUMSQ);

    k_proto<<<NC, 256, 0, stream>>>(ws + OFF_MUS, ws + OFF_CC, ws + OFF_P, ws + OFF_PNORM);
    k_rownorm<<<NB, 256, 0, stream>>>(xq, ws + OFF_QNORM);
    k_var<<<NB, 256, 0, stream>>>(xq, yq, ws + OFF_SUMSQ, ws + OFF_CQ, ws + OFF_D2R);

    k_wmma_logit<<<64, 256, 0, stream>>>(xq, ws + OFF_P, ws + OFF_PNORM,
                                         ws + OFF_QNORM, ws + OFF_CC, ws + OFF_LOGIT);
    k_fixup<<<NB, 256, 0, stream>>>(xq, ws + OFF_MUS, ws + OFF_CC, ys, pos, ws + OFF_LOGIT);

    k_loss<<<NB, 256, 0, stream>>>(ws + OFF_LOGIT, ys, pos, ws + OFF_LOSSR);
    k_final<<<1, 256, 0, stream>>>(ws + OFF_LOSSR, ws + OFF_D2R, ws + OFF_CQ, out);
}